// DeepSeekMoE_10290741641431
// MI455X (gfx1250) — compile-verified
//
#include <hip/hip_runtime.h>
#include <math.h>

// ---------------- problem constants (fixed shapes from reference) ----------
#define D_DIM 1024        // hidden dim
#define N_TOK 8192        // B*S tokens
#define E_NUM 7           // experts
#define CAPC  2574        // int(N * top_k / E * 1.1)
#define CPAD  2688        // capacity rounded up to 128-row tiles (21*128)

// ---------------- GEMM tiling ----------------------------------------------
#define BM 128
#define BN 128
#define BK 16
#define PAD 20            // LDS row pitch in floats: even (v2f) + 16B-aligned

typedef __attribute__((ext_vector_type(2))) float v2f;
typedef __attribute__((ext_vector_type(8))) float v8f;

__device__ __forceinline__ v8f wmma_f32(v2f a, v2f b, v8f c) {
  // V_WMMA_F32_16X16X4_F32 : D = A(16x4) x B(4x16) + C(16x16), all f32
  return __builtin_amdgcn_wmma_f32_16x16x4_f32(false, a, false, b,
                                               (short)0, c, false, false);
}

// ============================================================================
// Generic 128x128x(K=1024) fp32 WMMA GEMM:  Out = act(A @ W + bias)
//   GIN : A rows gathered through token index table (expert up-projection)
//   SOUT: output rows scattered with atomicAdd, scaled by gate (expert down)
//   RELU: apply relu in epilogue
// Double-buffered LDS software pipeline: 1 barrier per K-stage, global loads
// run 2 stages ahead of the WMMA stream.
// ============================================================================
template<bool GIN, bool SOUT, bool RELU>
__global__ __launch_bounds__(256)
void gemm_kernel(const float* __restrict__ Abase, const float* __restrict__ Wbase,
                 const float* __restrict__ biasBase, float* __restrict__ Outbase,
                 const int* __restrict__ tokBase, const float* __restrict__ gateBase,
                 int M, long aZ, long wZ, long bZ, long oZ, int tokZ)
{
  __shared__ float As[2][BM * PAD];   // As[buf][row*PAD + k]  (128x16)
  __shared__ float Bs[2][BN * PAD];   // Bs[buf][col*PAD + k]  (transposed)
  __shared__ int   tokL[BM];
  __shared__ float gL[BM];

  const int e = blockIdx.z;
  const float* A    = Abase   + (size_t)e * aZ;
  const float* W    = Wbase   + (size_t)e * wZ;
  const float* bias = biasBase+ (size_t)e * bZ;
  float*       Out  = Outbase + (size_t)e * oZ;

  const int tid     = threadIdx.x;
  const int rowBase = blockIdx.x * BM;
  const int nBase   = blockIdx.y * BN;

  if (GIN || SOUT) {
    const int*   tok  = tokBase  + (size_t)e * tokZ;
    const float* gate = gateBase ? gateBase + (size_t)e * tokZ : nullptr;
    if (tid < BM) {
      int slot = rowBase + tid;
      tokL[tid] = (slot < tokZ) ? tok[slot] : N_TOK;
      if (SOUT) gL[tid] = (slot < tokZ) ? gate[slot] : 0.f;
    }
    __syncthreads();
  }

  // ---- global->LDS load mapping (256 threads, 8 floats each per tile) ----
  // A tile 128x16 : thread -> row = tid>>1, 8 contiguous k at (tid&1)*8
  const int aRow = tid >> 1;
  const int aCol = (tid & 1) * 8;
  const float* aPtr;
  if (GIN) {
    int t = tokL[aRow];
    aPtr = (t < N_TOK) ? (A + (size_t)t * D_DIM + aCol) : nullptr;
  } else {
    int gr = rowBase + aRow;
    aPtr = (gr < M) ? (A + (size_t)gr * D_DIM + aCol) : nullptr;
  }
  // B tile 16x128 : thread -> krow = tid>>4, 8 contiguous n at (tid&15)*8
  const int bK   = tid >> 4;
  const int bCol = (tid & 15) * 8;
  const float* wPtr = W + (size_t)bK * D_DIM + nBase + bCol;

  // ---- wave/WMMA mapping: 8 waves -> 4 row groups x 2 col groups ---------
  const int lane  = tid & 31;
  const int wave  = tid >> 5;
  const int wr    = wave & 3;            // row group: 32 rows each
  const int wc    = wave >> 2;           // col group: 64 cols each
  const int lrow  = lane & 15;           // M (for A) / N (for B,C) index
  const int khalf = (lane >> 4) << 1;    // K pair: lanes 16-31 hold K+2

  v8f acc[2][4];
  #pragma unroll
  for (int r = 0; r < 2; ++r)
    #pragma unroll
    for (int c = 0; c < 4; ++c) {
      v8f z = {0.f,0.f,0.f,0.f,0.f,0.f,0.f,0.f};
      acc[r][c] = z;
    }

  const float4 f4z = make_float4(0.f, 0.f, 0.f, 0.f);

#define GLOAD(KB)                                                              \
  do {                                                                         \
    if (aPtr) {                                                                \
      sA0 = *(const float4*)(aPtr + (KB));                                     \
      sA1 = *(const float4*)(aPtr + (KB) + 4);                                 \
    } else { sA0 = f4z; sA1 = f4z; }                                           \
    sB0 = *(const float4*)(wPtr + (size_t)(KB) * D_DIM);                       \
    sB1 = *(const float4*)(wPtr + (size_t)(KB) * D_DIM + 4);                   \
  } while (0)

#define LSTORE(BUF)                                                            \
  do {                                                                         \
    *(float4*)&As[BUF][aRow * PAD + aCol]     = sA0;                           \
    *(float4*)&As[BUF][aRow * PAD + aCol + 4] = sA1;                           \
    Bs[BUF][(bCol+0)*PAD + bK] = sB0.x;  Bs[BUF][(bCol+1)*PAD + bK] = sB0.y;   \
    Bs[BUF][(bCol+2)*PAD + bK] = sB0.z;  Bs[BUF][(bCol+3)*PAD + bK] = sB0.w;   \
    Bs[BUF][(bCol+4)*PAD + bK] = sB1.x;  Bs[BUF][(bCol+5)*PAD + bK] = sB1.y;   \
    Bs[BUF][(bCol+6)*PAD + bK] = sB1.z;  Bs[BUF][(bCol+7)*PAD + bK] = sB1.w;   \
  } while (0)

#define COMPUTE(BUF)                                                           \
  do {                                                                         \
    _Pragma("unroll")                                                          \
    for (int kk = 0; kk < BK; kk += 4) {                                       \
      v2f a0f = *(const v2f*)&As[BUF][(wr*32 +      lrow) * PAD + kk + khalf]; \
      v2f a1f = *(const v2f*)&As[BUF][(wr*32 + 16 + lrow) * PAD + kk + khalf]; \
      v2f bf0 = *(const v2f*)&Bs[BUF][(wc*64 +  0 + lrow) * PAD + kk + khalf]; \
      v2f bf1 = *(const v2f*)&Bs[BUF][(wc*64 + 16 + lrow) * PAD + kk + khalf]; \
      v2f bf2 = *(const v2f*)&Bs[BUF][(wc*64 + 32 + lrow) * PAD + kk + khalf]; \
      v2f bf3 = *(const v2f*)&Bs[BUF][(wc*64 + 48 + lrow) * PAD + kk + khalf]; \
      acc[0][0] = wmma_f32(a0f, bf0, acc[0][0]);                               \
      acc[1][0] = wmma_f32(a1f, bf0, acc[1][0]);                               \
      acc[0][1] = wmma_f32(a0f, bf1, acc[0][1]);                               \
      acc[1][1] = wmma_f32(a1f, bf1, acc[1][1]);                               \
      acc[0][2] = wmma_f32(a0f, bf2, acc[0][2]);                               \
      acc[1][2] = wmma_f32(a1f, bf2, acc[1][2]);                               \
      acc[0][3] = wmma_f32(a0f, bf3, acc[0][3]);                               \
      acc[1][3] = wmma_f32(a1f, bf3, acc[1][3]);                               \
    }                                                                          \
  } while (0)

  float4 sA0, sA1, sB0, sB1;            // global->LDS staging registers
  const int NT = D_DIM / BK;            // 64 K-stages (even)

  // pipeline prologue: buf0 <- stage 0, staging regs <- stage 1
  GLOAD(0);
  LSTORE(0);
  GLOAD(BK);
  __syncthreads();

  // steady state: 1 barrier per stage; global loads run 2 stages ahead
  for (int kt = 0; kt < NT - 2; kt += 2) {
    LSTORE(1);                                      // buf1 <- stage kt+1
    __builtin_prefetch(wPtr + (size_t)(kt + 4) * BK * D_DIM, 0, 1);
    GLOAD((kt + 2) * BK);                           // regs <- stage kt+2
    COMPUTE(0);                                     // consume stage kt
    __syncthreads();
    LSTORE(0);                                      // buf0 <- stage kt+2
    GLOAD((kt + 3) * BK);                           // regs <- stage kt+3
    COMPUTE(1);                                     // consume stage kt+1
    __syncthreads();
  }
  // epilogue stages NT-2 (in buf0) and NT-1 (in staging regs)
  LSTORE(1);
  COMPUTE(0);
  __syncthreads();
  COMPUTE(1);

#undef GLOAD
#undef LSTORE
#undef COMPUTE

  // ---- epilogue -----------------------------------------------------------
  float bv[4];
  #pragma unroll
  for (int c = 0; c < 4; ++c)
    bv[c] = bias[nBase + wc*64 + c*16 + lrow];

  const int laneHi = (lane >> 4) << 3;   // C layout: upper half lanes hold M+8
  #pragma unroll
  for (int r = 0; r < 2; ++r) {
    #pragma unroll
    for (int c = 0; c < 4; ++c) {
      const int col = nBase + wc*64 + c*16 + lrow;
      #pragma unroll
      for (int vv = 0; vv < 8; ++vv) {
        float v = acc[r][c][vv] + bv[c];
        if (RELU) v = fmaxf(v, 0.f);
        int rowIn = wr*32 + r*16 + vv + laneHi;
        if (SOUT) {
          int slot = rowBase + rowIn;
          int t    = tokL[rowIn];
          if (slot < CAPC && t < N_TOK)
            atomicAdd(&Out[(size_t)t * D_DIM + col], v * gL[rowIn]);
        } else {
          int gr = rowBase + rowIn;
          if (gr < M) Out[(size_t)gr * D_DIM + col] = v;
        }
      }
    }
  }
}

// ============================================================================
// Router: noisy top-2 gating. One wave per token (8 tokens / 256-thread block)
// ============================================================================
__global__ __launch_bounds__(256)
void router_kernel(const float* __restrict__ x, const float* __restrict__ noise,
                   const float* __restrict__ Wr, const float* __restrict__ br,
                   const float* __restrict__ Wn, const float* __restrict__ bn,
                   unsigned char* __restrict__ mask, float* __restrict__ gfull,
                   float* __restrict__ sumP, float* __restrict__ sumM)
{
  __shared__ float WrL[D_DIM * E_NUM];
  __shared__ float WnL[D_DIM * E_NUM];
  __shared__ float bp[E_NUM], bm[E_NUM];

  const int tid = threadIdx.x;
  for (int i = tid; i < D_DIM * E_NUM; i += 256) { WrL[i] = Wr[i]; WnL[i] = Wn[i]; }
  if (tid < E_NUM) { bp[tid] = 0.f; bm[tid] = 0.f; }
  __syncthreads();

  const int lane = tid & 31, wave = tid >> 5;
  const int t = blockIdx.x * 8 + wave;
  const float* xr = x + (size_t)t * D_DIM;

  float aR[E_NUM], aN[E_NUM];
  #pragma unroll
  for (int e = 0; e < E_NUM; ++e) { aR[e] = 0.f; aN[e] = 0.f; }
  for (int d = lane; d < D_DIM; d += 32) {
    float xv = xr[d];
    #pragma unroll
    for (int e = 0; e < E_NUM; ++e) {
      aR[e] += xv * WrL[d * E_NUM + e];
      aN[e] += xv * WnL[d * E_NUM + e];
    }
  }
  #pragma unroll
  for (int e = 0; e < E_NUM; ++e)
    for (int off = 16; off > 0; off >>= 1) {
      aR[e] += __shfl_xor(aR[e], off, 32);
      aN[e] += __shfl_xor(aN[e], off, 32);
    }

  if (lane == 0) {
    float nz[E_NUM];
    #pragma unroll
    for (int e = 0; e < E_NUM; ++e) {
      float lg = aR[e] + br[e];
      float z  = aN[e] + bn[e];
      float sp = fmaxf(z, 0.f) + log1pf(expf(-fabsf(z)));   // stable softplus
      nz[e] = lg + noise[(size_t)t * E_NUM + e] * sp;
    }
    int i1 = 0; float v1 = nz[0];
    #pragma unroll
    for (int e = 1; e < E_NUM; ++e) if (nz[e] > v1) { v1 = nz[e]; i1 = e; }
    int i2 = -1; float v2 = -1e30f;
    #pragma unroll
    for (int e = 0; e < E_NUM; ++e) if (e != i1 && nz[e] > v2) { v2 = nz[e]; i2 = e; }
    float e2 = expf(v2 - v1);
    float inv = 1.f / (1.f + e2);
    float p1 = inv, p2 = e2 * inv;                          // softmax over top-2
    #pragma unroll
    for (int e = 0; e < E_NUM; ++e) {
      gfull[(size_t)t * E_NUM + e] = 0.f;
      mask[(size_t)e * N_TOK + t]  = 0;
    }
    gfull[(size_t)t * E_NUM + i1] = p1;
    gfull[(size_t)t * E_NUM + i2] = p2;
    mask[(size_t)i1 * N_TOK + t] = 1;
    mask[(size_t)i2 * N_TOK + t] = 1;
    atomicAdd(&bp[i1], p1); atomicAdd(&bp[i2], p2);
    atomicAdd(&bm[i1], 1.f); atomicAdd(&bm[i2], 1.f);
  }
  __syncthreads();
  if (tid < E_NUM) { atomicAdd(&sumP[tid], bp[tid]); atomicAdd(&sumM[tid], bm[tid]); }
}

// ============================================================================
// Dispatch: per-expert FCFS capacity scan (one wave per expert, ballot prefix)
// ============================================================================
__global__ void dispatch_kernel(const unsigned char* __restrict__ mask,
                                const float* __restrict__ gfull,
                                int* __restrict__ idxs, float* __restrict__ gs)
{
  const int lane = threadIdx.x & 31;
  const int e    = threadIdx.x >> 5;
  if (e >= E_NUM) return;

  unsigned cnt = 0;
  const unsigned ltmask = (1u << lane) - 1u;
  for (int base = 0; base < N_TOK; base += 32) {
    int  t = base + lane;
    bool m = mask[(size_t)e * N_TOK + t] != 0;
    unsigned b32 = (unsigned)__ballot(m);
    if (m) {
      unsigned slot = cnt + __popc(b32 & ltmask);
      if (slot < CAPC) {
        idxs[(size_t)e * CPAD + slot] = t;
        gs  [(size_t)e * CPAD + slot] = gfull[(size_t)t * E_NUM + e];
      }
    }
    cnt += __popc(b32);
  }
  unsigned start = cnt < CAPC ? cnt : (unsigned)CAPC;
  for (unsigned s = start + lane; s < CPAD; s += 32) {
    idxs[(size_t)e * CPAD + s] = N_TOK;    // padding slot
    gs  [(size_t)e * CPAD + s] = 0.f;
  }
}

__global__ void zero_kernel(float* sums) {
  if (threadIdx.x < 2 * E_NUM) sums[threadIdx.x] = 0.f;
}

__global__ void loss_kernel(const float* __restrict__ sums, float* __restrict__ out) {
  if (threadIdx.x == 0) {
    float acc = 0.f;
    for (int e = 0; e < E_NUM; ++e) acc += sums[e] * sums[E_NUM + e];
    out[(size_t)N_TOK * D_DIM] =
        (float)E_NUM * acc / ((float)N_TOK * (float)N_TOK);
  }
}

// ============================================================================
extern "C" void kernel_launch(void* const* d_in, const int* in_sizes, int n_in,
                              void* d_out, int out_size, void* d_ws, size_t ws_size,
                              hipStream_t stream)
{
  const float* x    = (const float*)d_in[0];
  const float* noise= (const float*)d_in[1];
  const float* Wr   = (const float*)d_in[2];
  const float* br   = (const float*)d_in[3];
  const float* Wn   = (const float*)d_in[4];
  const float* bn   = (const float*)d_in[5];
  const float* W1   = (const float*)d_in[6];
  const float* b1   = (const float*)d_in[7];
  const float* W2   = (const float*)d_in[8];
  const float* b2   = (const float*)d_in[9];
  const float* Ws1  = (const float*)d_in[10];
  const float* bs1  = (const float*)d_in[11];
  const float* Ws2  = (const float*)d_in[12];
  const float* bs2  = (const float*)d_in[13];
  float* out = (float*)d_out;

  char* ws = (char*)d_ws;
  size_t off = 0;
  auto alloc = [&](size_t bytes) {
    size_t p = off; off += (bytes + 255) & ~(size_t)255; return p;
  };
  unsigned char* mask = (unsigned char*)(ws + alloc((size_t)E_NUM * N_TOK));
  float* gfull = (float*)(ws + alloc((size_t)N_TOK * E_NUM * 4));
  int*   idxs  = (int*)  (ws + alloc((size_t)E_NUM * CPAD * 4));
  float* gs    = (float*)(ws + alloc((size_t)E_NUM * CPAD * 4));
  float* sums  = (float*)(ws + alloc(2 * E_NUM * 4));
  float* Hs    = (float*)(ws + alloc((size_t)N_TOK * D_DIM * 4));
  float* He    = (float*)(ws + alloc((size_t)E_NUM * CPAD * D_DIM * 4));

  zero_kernel<<<1, 64, 0, stream>>>(sums);
  router_kernel<<<N_TOK / 8, 256, 0, stream>>>(x, noise, Wr, br, Wn, bn,
                                               mask, gfull, sums, sums + E_NUM);
  dispatch_kernel<<<1, 224, 0, stream>>>(mask, gfull, idxs, gs);

  dim3 gShared(N_TOK / BM, D_DIM / BN, 1);
  // shared expert: Hs = relu(x @ Ws1 + bs1) ; out = Hs @ Ws2 + bs2
  gemm_kernel<false,false,true ><<<gShared, 256, 0, stream>>>(
      x,  Ws1, bs1, Hs,  nullptr, nullptr, N_TOK, 0, 0, 0, 0, CPAD);
  gemm_kernel<false,false,false><<<gShared, 256, 0, stream>>>(
      Hs, Ws2, bs2, out, nullptr, nullptr, N_TOK, 0, 0, 0, 0, CPAD);

  dim3 gExp(CPAD / BM, D_DIM / BN, E_NUM);
  // experts: He = relu(gather(x) @ W1[e] + b1[e]) ; out += gate * (He @ W2[e] + b2[e])
  gemm_kernel<true ,false,true ><<<gExp, 256, 0, stream>>>(
      x,  W1, b1, He,  idxs, nullptr, CPAD,
      0, (long)D_DIM * D_DIM, D_DIM, (long)CPAD * D_DIM, CPAD);
  gemm_kernel<false,true ,false><<<gExp, 256, 0, stream>>>(
      He, W2, b2, out, idxs, gs, CPAD,
      (long)CPAD * D_DIM, (long)D_DIM * D_DIM, D_DIM, 0, CPAD);

  loss_kernel<<<1, 32, 0, stream>>>(sums, out);
}